// MattingSystem_8985071583323
// MI455X (gfx1250) — compile-verified
//
#include <hip/hip_runtime.h>

typedef float v2f __attribute__((ext_vector_type(2)));
typedef float v4f __attribute__((ext_vector_type(4)));
typedef float v8f __attribute__((ext_vector_type(8)));
typedef int   v4i __attribute__((ext_vector_type(4)));

#if __has_builtin(__builtin_amdgcn_global_load_async_to_lds_b128)
#define HAVE_ASYNC_LDS 1
#else
#define HAVE_ASYNC_LDS 0
#endif

// ---------------------------------------------------------------------------
// Utility: zero a float buffer (harness poisons d_ws with 0xAA)
// ---------------------------------------------------------------------------
__global__ void zero_f32(float* __restrict__ p, long long count) {
  long long i = (long long)blockIdx.x * blockDim.x + threadIdx.x;
  long long stride = (long long)gridDim.x * blockDim.x;
  for (; i < count; i += stride) p[i] = 0.0f;
}

// ---------------------------------------------------------------------------
// Scatter Wcm (COO, duplicates add) scaled by CM_weights[row] into dense W.
// ---------------------------------------------------------------------------
__global__ void scatter_wcm(const int* __restrict__ rows,
                            const int* __restrict__ cols,
                            const float* __restrict__ data,
                            const float* __restrict__ cmw,
                            float* __restrict__ W, int nnz, int n) {
  int e = blockIdx.x * blockDim.x + threadIdx.x;
  if (e >= nnz) return;
  int r = rows[e];
  int c = cols[e];
  atomicAdd(&W[(long long)r * n + c], data[e] * cmw[r]);
}

// ---------------------------------------------------------------------------
// In-place: W  ->  L = diag(rowsum(W)) - W.   One block per row.
// ---------------------------------------------------------------------------
__global__ void form_laplacian(float* __restrict__ W, int n) {
  __shared__ float red[256];
  const int r = blockIdx.x;
  const long long base = (long long)r * n;
  float s = 0.0f;
  for (int c = threadIdx.x; c < n; c += blockDim.x) s += W[base + c];
  red[threadIdx.x] = s;
  __syncthreads();
  for (int off = 128; off > 0; off >>= 1) {
    if (threadIdx.x < off) red[threadIdx.x] += red[threadIdx.x + off];
    __syncthreads();
  }
  const float rowsum = red[0];
  for (int c = threadIdx.x; c < n; c += blockDim.x) {
    float w = W[base + c];
    float v = -w;
    if (c == r) v += rowsum;
    W[base + c] = v;
  }
}

// ---------------------------------------------------------------------------
// Laplacian-of-symmetrized-scatter directly into A (after GEMM wrote A):
//   entry (r,c,v):  A[r,r]+=v/2  A[c,c]+=v/2  A[r,c]-=v/2  A[c,r]-=v/2
// ---------------------------------------------------------------------------
__device__ __forceinline__ void lap_scatter(float* A, int n, int r, int c, float halfv) {
  atomicAdd(&A[(long long)r * n + r],  halfv);
  atomicAdd(&A[(long long)c * n + c],  halfv);
  atomicAdd(&A[(long long)r * n + c], -halfv);
  atomicAdd(&A[(long long)c * n + r], -halfv);
}

__global__ void scatter_loc(const int* __restrict__ inInd,
                            const float* __restrict__ flows,  // (9,9,M) flat
                            const float* __restrict__ locw,
                            const int* __restrict__ widthp,
                            float* __restrict__ A, int mloc, int n) {
  int t = blockIdx.x * blockDim.x + threadIdx.x;
  if (t >= mloc * 9) return;
  const int m = t % mloc;
  const int j = t / mloc;            // 0..8
  const int w = widthp[0];
  const int ind = inInd[m];
  const int a = (j / 3) - 1;         // offs = a + b*w
  const int bq = (j % 3) - 1;
  const int r = ind - 1 - w;         // neigh[:,0] broadcast as row
  const int c = ind + a + bq * w;
  // vals0[m][j] = LOC_flows[j,0,m] * LOC_weights[ind]
  const float v = 0.5f * flows[(long long)j * 9 * mloc + m] * locw[ind];
  lap_scatter(A, n, r, c, v);
}

__global__ void scatter_iu(const int* __restrict__ inInd,
                           const int* __restrict__ neighInd,  // (M,5)
                           const float* __restrict__ flows,   // (M,5)
                           const float* __restrict__ iuw,
                           float* __restrict__ A, int miu, int n) {
  int t = blockIdx.x * blockDim.x + threadIdx.x;
  if (t >= miu * 5) return;
  const int m = t / 5;
  const int j = t % 5;
  const int r = inInd[m];
  const int c = neighInd[m * 5 + j];
  const float v = 0.5f * flows[m * 5 + j] * iuw[r];
  lap_scatter(A, n, r, c, v);
}

// ---------------------------------------------------------------------------
// Diagonal confidence terms + RHS b.
// ---------------------------------------------------------------------------
__global__ void diag_and_b(const float* __restrict__ kuw,
                           const float* __restrict__ ktouconf,
                           const float* __restrict__ known,
                           const float* __restrict__ ktou,
                           const float* __restrict__ lmbda,
                           float* __restrict__ A, float* __restrict__ b, int n) {
  int i = blockIdx.x * blockDim.x + threadIdx.x;
  if (i >= n) return;
  const float d = kuw[i] * ktouconf[i] + lmbda[0] * known[i];
  atomicAdd(&A[(long long)i * n + i], d);
  b[i] = d * ktou[i];
}

// ---------------------------------------------------------------------------
// A = L^T * L  via V_WMMA_F32_16X16X4_F32, double-buffered K-slabs in LDS.
// Block: 256 threads = 8 waves (2x4). Wave tile 32x32 (2x2 accumulators).
// Block tile 64(M) x 128(N); K staged 16 rows at a time.
// f32 WMMA frag layout (ISA 7.12.2): lanes 0-15 -> K=0(v0),K=1(v1);
// lanes 16-31 -> K=2(v0),K=3(v1); column = lane&15.
// Staging uses GLOBAL_LOAD_ASYNC_TO_LDS_B128 (ASYNCcnt) when the toolchain
// exposes the builtin; otherwise a vectorized sync copy.
// ---------------------------------------------------------------------------
#define KT 16
#define PA 68    // padded pitch for 64-wide slab (bank-conflict free, 16B-aligned rows)
#define PB 132   // padded pitch for 128-wide slab

#if HAVE_ASYNC_LDS
// Per the compile probe, param 0 is `v4i addrspace(1)*` (AS1 prints as
// __device__ in HIP diagnostics); param 1 is the LDS-side pointer.
typedef __attribute__((address_space(1))) v4i* gas_b128_t;
typedef __attribute__((address_space(3))) v4i* lds_b128_t;

__device__ __forceinline__ void async_b128(const float* g, float* l) {
  __builtin_amdgcn_global_load_async_to_lds_b128(
      (gas_b128_t)(void*)const_cast<float*>(g), (lds_b128_t)(void*)l, 0, 0);
}

template <int N>
__device__ __forceinline__ void wait_async() {
#if __has_builtin(__builtin_amdgcn_s_wait_asynccnt)
  __builtin_amdgcn_s_wait_asynccnt(N);
#else
  asm volatile("s_wait_asynccnt %0" ::"i"(N) : "memory");
#endif
}
#endif  // HAVE_ASYNC_LDS

__global__ __launch_bounds__(256) void gemm_ltl_wmma(const float* __restrict__ L,
                                                     float* __restrict__ A, int n) {
  __shared__ float sA[2][KT * PA];
  __shared__ float sB[2][KT * PB];

  const int tid  = threadIdx.x;
  const int lane = tid & 31;
  const int wave = tid >> 5;
  const int wm   = wave >> 2;   // 0..1 -> 32-row sub-tile
  const int wn   = wave & 3;    // 0..3 -> 32-col sub-tile
  const int half = lane >> 4;   // lane half selects K=0,1 vs K=2,3
  const int l16  = lane & 15;

  const int rowBase = blockIdx.y * 64;    // M tile (columns of L feeding A-op)
  const int colBase = blockIdx.x * 128;   // N tile (columns of L feeding B-op)

  // Per-thread staging chunks (b128 = 4 floats each):
  //  slabA: 256 chunks -> 1/thread;  slabB: 512 chunks -> 2/thread.
  const int akr = tid >> 4, ac4 = (tid & 15) * 4;
  const int bkr0 = tid >> 5,          bc40 = (tid & 31) * 4;
  const int bkr1 = (tid + 256) >> 5,  bc41 = bc40;  // chunk tid+256: rows +8

  v8f acc[2][2] = {};

  const int am0 = wm * 32 + l16;
  const int am1 = am0 + 16;
  const int bn0 = wn * 32 + l16;
  const int bn1 = bn0 + 16;

  const int iters = n / KT;

#if HAVE_ASYNC_LDS
  // ---- async double-buffered pipeline (3 b128 asyncs per thread per slab) ---
  {
    const long long kk = 0;
    async_b128(&L[(kk + akr) * n + rowBase + ac4],   &sA[0][akr * PA + ac4]);
    async_b128(&L[(kk + bkr0) * n + colBase + bc40], &sB[0][bkr0 * PB + bc40]);
    async_b128(&L[(kk + bkr1) * n + colBase + bc41], &sB[0][bkr1 * PB + bc41]);
  }
  for (int it = 0; it < iters; ++it) {
    const int buf = it & 1;
    const bool pref = (it + 1) < iters;
    if (pref) {
      const long long kk = (long long)(it + 1) * KT;
      async_b128(&L[(kk + akr) * n + rowBase + ac4],   &sA[buf ^ 1][akr * PA + ac4]);
      async_b128(&L[(kk + bkr0) * n + colBase + bc40], &sB[buf ^ 1][bkr0 * PB + bc40]);
      async_b128(&L[(kk + bkr1) * n + colBase + bc41], &sB[buf ^ 1][bkr1 * PB + bc41]);
      wait_async<3>();   // async loads complete in order: slab `it` is resident
    } else {
      wait_async<0>();
    }
    __syncthreads();
#else
  // ---- sync fallback: vectorized b128 staging, single-phase per slab -------
  for (int it = 0; it < iters; ++it) {
    const int buf = it & 1;
    {
      const long long kk = (long long)it * KT;
      v4f va  = *(const v4f*)&L[(kk + akr) * n + rowBase + ac4];
      v4f vb0 = *(const v4f*)&L[(kk + bkr0) * n + colBase + bc40];
      v4f vb1 = *(const v4f*)&L[(kk + bkr1) * n + colBase + bc41];
      *(v4f*)&sA[buf][akr * PA + ac4]   = va;
      *(v4f*)&sB[buf][bkr0 * PB + bc40] = vb0;
      *(v4f*)&sB[buf][bkr1 * PB + bc41] = vb1;
    }
    __syncthreads();
#endif

#pragma unroll
    for (int ks = 0; ks < KT; ks += 4) {
      const int k0 = ks + 2 * half;
      v2f a0, a1, b0, b1;
      a0.x = sA[buf][k0 * PA + am0];       a0.y = sA[buf][(k0 + 1) * PA + am0];
      a1.x = sA[buf][k0 * PA + am1];       a1.y = sA[buf][(k0 + 1) * PA + am1];
      b0.x = sB[buf][k0 * PB + bn0];       b0.y = sB[buf][(k0 + 1) * PB + bn0];
      b1.x = sB[buf][k0 * PB + bn1];       b1.y = sB[buf][(k0 + 1) * PB + bn1];

      acc[0][0] = __builtin_amdgcn_wmma_f32_16x16x4_f32(false, a0, false, b0,
                                                        (short)0, acc[0][0], false, false);
      acc[0][1] = __builtin_amdgcn_wmma_f32_16x16x4_f32(false, a0, false, b1,
                                                        (short)0, acc[0][1], false, false);
      acc[1][0] = __builtin_amdgcn_wmma_f32_16x16x4_f32(false, a1, false, b0,
                                                        (short)0, acc[1][0], false, false);
      acc[1][1] = __builtin_amdgcn_wmma_f32_16x16x4_f32(false, a1, false, b1,
                                                        (short)0, acc[1][1], false, false);
    }
    __syncthreads();
  }

  // Writeback (plain store: GEMM runs first, scatters are applied afterwards).
  // C/D layout -> VGPR r holds (M=r, N=l16) for lanes 0-15, (M=r+8, N=l16)
  // for lanes 16-31.
#pragma unroll
  for (int i = 0; i < 2; ++i) {
#pragma unroll
    for (int j = 0; j < 2; ++j) {
      const int colg = colBase + wn * 32 + j * 16 + l16;
#pragma unroll
      for (int r = 0; r < 8; ++r) {
        const int rowg = rowBase + wm * 32 + i * 16 + r + 8 * half;
        A[(long long)rowg * n + colg] = acc[i][j][r];
      }
    }
  }
}

// ---------------------------------------------------------------------------
// Host orchestration (all on `stream`, graph-capture safe).
// Inputs (setup_inputs order):
//  0 height 1 width 2 CM_weights 3 LOC_weights 4 IU_weights 5 KU_weights
//  6 lmbda 7 kToUconf 8 known 9 kToU 10 Wcm_row 11 Wcm_col 12 Wcm_data
//  13 LOC_inInd 14 LOC_flows 15 IU_inInd 16 IU_flows 17 IU_neighInd
// ---------------------------------------------------------------------------
extern "C" void kernel_launch(void* const* d_in, const int* in_sizes, int n_in,
                              void* d_out, int out_size, void* d_ws, size_t ws_size,
                              hipStream_t stream) {
  const int n    = in_sizes[2];    // N = H*W = 4096
  const int nnz  = in_sizes[10];   // 20*N
  const int mloc = in_sizes[13];   // M_LOC
  const int miu  = in_sizes[15];   // M_IU

  const int*   widthp      = (const int*)  d_in[1];
  const float* CM_weights  = (const float*)d_in[2];
  const float* LOC_weights = (const float*)d_in[3];
  const float* IU_weights  = (const float*)d_in[4];
  const float* KU_weights  = (const float*)d_in[5];
  const float* lmbda       = (const float*)d_in[6];
  const float* kToUconf    = (const float*)d_in[7];
  const float* known       = (const float*)d_in[8];
  const float* kToU        = (const float*)d_in[9];
  const int*   Wcm_row     = (const int*)  d_in[10];
  const int*   Wcm_col     = (const int*)  d_in[11];
  const float* Wcm_data    = (const float*)d_in[12];
  const int*   LOC_inInd   = (const int*)  d_in[13];
  const float* LOC_flows   = (const float*)d_in[14];
  const int*   IU_inInd    = (const int*)  d_in[15];
  const float* IU_flows    = (const float*)d_in[16];
  const int*   IU_neighInd = (const int*)  d_in[17];

  float* A  = (float*)d_out;                      // N x N
  float* b  = A + (long long)n * n;               // N
  float* Lm = (float*)d_ws;                       // dense Lcm scratch, N*N f32

  const long long nn = (long long)n * n;

  // 1) zero the L scratch only (A is fully overwritten by the GEMM)
  zero_f32<<<2048, 256, 0, stream>>>(Lm, nn);

  // 2) Wcm scatter (row-scaled), then L = diag(rowsum) - W in place
  scatter_wcm<<<(nnz + 255) / 256, 256, 0, stream>>>(Wcm_row, Wcm_col, Wcm_data,
                                                     CM_weights, Lm, nnz, n);
  form_laplacian<<<n, 256, 0, stream>>>(Lm, n);

  // 3) A = L^T L  (dense f32 WMMA GEMM; L fits in the 192MB L2)
  dim3 grid(n / 128, n / 64);
  gemm_ltl_wmma<<<grid, 256, 0, stream>>>(Lm, A, n);

  // 4) LOC / IU symmetrized Laplacians scattered on top of A
  scatter_loc<<<(mloc * 9 + 255) / 256, 256, 0, stream>>>(LOC_inInd, LOC_flows,
                                                          LOC_weights, widthp,
                                                          A, mloc, n);
  scatter_iu<<<(miu * 5 + 255) / 256, 256, 0, stream>>>(IU_inInd, IU_neighInd,
                                                        IU_flows, IU_weights,
                                                        A, miu, n);

  // 5) diagonal terms + b
  diag_and_b<<<(n + 255) / 256, 256, 0, stream>>>(KU_weights, kToUconf, known,
                                                  kToU, lmbda, A, b, n);
}